// PermDistDissconnected_34376918237466
// MI455X (gfx1250) — compile-verified
//
#include <hip/hip_runtime.h>
#include <math.h>

// ---------------------------------------------------------------------------
// PermDistDisconnected: 63 independent MLPs (5x [1024x1024] + [64x1024] + softmax)
// Bandwidth-bound: 1.32 GB of weights read once -> ~57us floor at 23.3 TB/s.
// Hidden layers via V_WMMA_F32_16X16X4_F32 (wave32). B = x broadcast across
// all 16 columns so lanes j and 16+j together hold the full 16-row result slice.
// K-chunks remapped so each lane does one b128 load feeding TWO wmmas:
//   h=0 lanes own k = kb..kb+3, h=1 lanes own k = kb+4..kb+7 (sum over k is
//   commutative; B supplies the matching x[k] per half-wave).
// ---------------------------------------------------------------------------

typedef float v2f __attribute__((ext_vector_type(2)));
typedef float v4f __attribute__((ext_vector_type(4)));
typedef float v8f __attribute__((ext_vector_type(8)));

#define NMODELS 63
#define LSZ     1024
#define NHID    5
#define CLAMPV  1e9f

__global__ void pdd_init_ones(float* __restrict__ p, int n) {
    int i = blockIdx.x * blockDim.x + threadIdx.x;
    if (i < n) p[i] = 1.0f;
}

// grid: 63 models * 8 blocks, block: 256 threads (8 waves, 16 rows/wave)
__global__ void pdd_hidden_layer(const float* __restrict__ Wh,  // [63,5,1024,1024]
                                 const float* __restrict__ bh,  // [63,5,1024]
                                 const float* __restrict__ xin, // [63,1024]
                                 float* __restrict__ xout,      // [63,1024]
                                 int layer) {
    __shared__ float xs[LSZ];

    const int model      = blockIdx.x >> 3;
    const int blkInModel = blockIdx.x & 7;

    // stage this model's activation vector in LDS
    for (int i = threadIdx.x; i < LSZ; i += blockDim.x)
        xs[i] = xin[model * LSZ + i];
    __syncthreads();

    const int wave = threadIdx.x >> 5;
    const int lane = threadIdx.x & 31;
    const int rowBase = (blkInModel * 8 + wave) * 16;   // 16 output rows per wave

    // A-matrix 16x4 f32 layout: lanes 0-15 supply K pair {0,1} of a wmma,
    // lanes 16-31 supply K pair {2,3}; lane & 15 selects the row.
    const int r = lane & 15;
    const int h = lane >> 4;

    const float* __restrict__ W =
        Wh + ((((size_t)model * NHID + layer) * LSZ) + (rowBase + r)) * LSZ;

    v8f c = {};
#pragma unroll 4
    for (int kb = 0; kb < LSZ; kb += 8) {
        // One 16B load per lane: h=0 lanes take k=kb..kb+3, h=1 lanes take
        // k=kb+4..kb+7 -> wave reads 16 rows x 32 contiguous bytes.
        // NT hint: weights are streamed exactly once, don't churn the 192MB L2.
        v4f a = __builtin_nontemporal_load((const v4f*)(W + kb + 4 * h));
        // Matching x values for this lane's k-range (one ds_load_b128).
        v4f xv = *(const v4f*)(xs + kb + 4 * h);

        v2f a1; a1.x = a.x; a1.y = a.y;
        v2f b1; b1.x = xv.x; b1.y = xv.y;
        v2f a2; a2.x = a.z; a2.y = a.w;
        v2f b2; b2.x = xv.z; b2.y = xv.w;

        // D = A*B + C  (8 args: neg_a, A, neg_b, B, c_mod, C, reuse_a, reuse_b)
        c = __builtin_amdgcn_wmma_f32_16x16x4_f32(false, a1, false, b1,
                                                  (short)0, c, false, false);
        c = __builtin_amdgcn_wmma_f32_16x16x4_f32(false, a2, false, b2,
                                                  (short)0, c, false, false);
    }

    // With B broadcast, C[m,n] == y[m] for every n: lane j (0-15) holds
    // y[rowBase+v] in c[v]; lane 16+j holds y[rowBase+8+v]. Lanes 0 and 16 store.
    if ((lane & 15) == 0) {
        const float* __restrict__ bias =
            bh + ((size_t)model * NHID + layer) * LSZ + rowBase + h * 8;
        float* __restrict__ out = xout + model * LSZ + rowBase + h * 8;
#pragma unroll
        for (int v = 0; v < 8; ++v) {
            float y = c[v] + bias[v];
            y = fminf(fmaxf(y, -CLAMPV), CLAMPV);   // clip
            out[v] = fmaxf(y, 0.0f);                // relu
        }
    }
}

// grid: 63 blocks (one per model), block: 256 threads (4 threads per logit)
__global__ void pdd_final_softmax(const float* __restrict__ Wf, // [63,64,1024]
                                  const float* __restrict__ bf, // [63,64]
                                  const float* __restrict__ x,  // [63,1024]
                                  float* __restrict__ out) {    // [63,64]
    __shared__ float xs[LSZ];
    __shared__ float part[64][4];
    __shared__ float logits[64];
    __shared__ float red[2];

    const int m = blockIdx.x;
    const int t = threadIdx.x;
    const int n = t >> 2;       // logit index 0..63
    const int sub = t & 3;      // quarter of the dot product

    for (int i = t; i < LSZ; i += blockDim.x) xs[i] = x[m * LSZ + i];
    __syncthreads();

    const float* __restrict__ w = Wf + ((size_t)m * 64 + n) * LSZ + sub * 256;
    const float* xp = xs + sub * 256;
    float s = 0.0f;
#pragma unroll 4
    for (int j = 0; j < 256; j += 4) {
        v4f wv = __builtin_nontemporal_load((const v4f*)(w + j));
        v4f xv = *(const v4f*)(xp + j);
        s += wv.x * xv.x + wv.y * xv.y + wv.z * xv.z + wv.w * xv.w;
    }
    part[n][sub] = s;
    __syncthreads();

    const int nd = 64 - m;  // valid out-dim for model m
    if (sub == 0) {
        float l = part[n][0] + part[n][1] + part[n][2] + part[n][3] + bf[m * 64 + n];
        logits[n] = (n < nd) ? l : -INFINITY;   // T = 1.0
    }
    __syncthreads();

    if (t == 0) {
        float mx = -INFINITY;
        for (int i = 0; i < nd; ++i) mx = fmaxf(mx, logits[i]);
        float sum = 0.0f;
        for (int i = 0; i < nd; ++i) sum += __expf(logits[i] - mx);
        red[0] = mx;
        red[1] = sum;
    }
    __syncthreads();

    if (t < 64) {
        float p = (t < nd) ? __expf(logits[t] - red[0]) / red[1] : 0.0f;
        out[m * 64 + t] = p;
    }
}

extern "C" void kernel_launch(void* const* d_in, const int* in_sizes, int n_in,
                              void* d_out, int out_size, void* d_ws, size_t ws_size,
                              hipStream_t stream) {
    const float* Wh = (const float*)d_in[0];  // [63,5,1024,1024]
    const float* bh = (const float*)d_in[1];  // [63,5,1024]
    const float* Wf = (const float*)d_in[2];  // [63,64,1024]
    const float* bf = (const float*)d_in[3];  // [63,64]
    // d_in[4] = mask: recomputed on device from model index, not needed.

    float* buf0 = (float*)d_ws;
    float* buf1 = buf0 + NMODELS * LSZ;

    const int nvec = NMODELS * LSZ;
    pdd_init_ones<<<(nvec + 255) / 256, 256, 0, stream>>>(buf0, nvec);

    float* cur = buf0;
    float* nxt = buf1;
    for (int layer = 0; layer < NHID; ++layer) {
        pdd_hidden_layer<<<NMODELS * 8, 256, 0, stream>>>(Wh, bh, cur, nxt, layer);
        float* tmp = cur; cur = nxt; nxt = tmp;
    }

    pdd_final_softmax<<<NMODELS, 256, 0, stream>>>(Wf, bf, cur, (float*)d_out);
}